// AnswerModule_38036230373461
// MI455X (gfx1250) — compile-verified
//
#include <hip/hip_runtime.h>
#include <math.h>

#define B_   32
#define QL_  64
#define PL_  4096
#define D2_  256
#define T_   4
#define MPAD 16   // padded M-rows per batch for the WMMA A operand

typedef __attribute__((ext_vector_type(2))) float v2f;
typedef __attribute__((ext_vector_type(8))) float v8f;

// ---------------------------------------------------------------- helpers
__device__ __forceinline__ float block_reduce_sum(float v, float* red) {
    int tid = threadIdx.x;
    red[tid] = v;
    __syncthreads();
    for (int s = 128; s > 0; s >>= 1) {
        if (tid < s) red[tid] += red[tid + s];
        __syncthreads();
    }
    float r = red[0];
    __syncthreads();
    return r;
}

// ---------------------------------------------------------------- s0 = attn(H_q, W_4)
__global__ void attn_s0_kernel(const float* __restrict__ Hq, const float* __restrict__ W4,
                               float* __restrict__ s_all) {
    int b = blockIdx.x, tid = threadIdx.x;
    __shared__ float lg[QL_];
    const float* Hqb = Hq + (size_t)b * QL_ * D2_;
    if (tid < QL_) {
        float acc = 0.f;
        for (int e = 0; e < D2_; ++e) acc += Hqb[tid * D2_ + e] * W4[e];
        lg[tid] = acc;
    }
    __syncthreads();
    if (tid == 0) {
        float m = -1e30f;
        for (int q = 0; q < QL_; ++q) m = fmaxf(m, lg[q]);
        float s = 0.f;
        for (int q = 0; q < QL_; ++q) { lg[q] = expf(lg[q] - m); s += lg[q]; }
        float inv = 1.f / s;
        for (int q = 0; q < QL_; ++q) lg[q] *= inv;
    }
    __syncthreads();
    float acc = 0.f;
    for (int q = 0; q < QL_; ++q) acc += lg[q] * Hqb[q * D2_ + tid];
    s_all[(size_t)b * D2_ + tid] = acc;   // row (t=0, b)
}

// ---------------------------------------------------------------- x = mean_n M[b,d,n]
__global__ void mean_x_kernel(const float* __restrict__ M, float* __restrict__ x) {
    int row = blockIdx.x;                       // b*256 + d
    const float* Mr = M + (size_t)row * PL_;
    __shared__ float red[256];
    float acc = 0.f;
    for (int n = threadIdx.x; n < PL_; n += 256) acc += Mr[n];
    float s = block_reduce_sum(acc, red);
    if (threadIdx.x == 0) x[row] = s * (1.0f / PL_);
}

// ---------------------------------------------------------------- gh = x @ W_hh^T + b_hh
__global__ void gh_kernel(const float* __restrict__ x, const float* __restrict__ Whh,
                          const float* __restrict__ bhh, float* __restrict__ gh) {
    int idx = blockIdx.x * blockDim.x + threadIdx.x;
    if (idx >= B_ * 3 * D2_) return;
    int b = idx / (3 * D2_), j = idx % (3 * D2_);
    const float* xb = x + (size_t)b * D2_;
    const float* wr = Whh + (size_t)j * D2_;
    float acc = bhh[j];
    for (int e = 0; e < D2_; ++e) acc += xb[e] * wr[e];
    gh[idx] = acc;
}

// ---------------------------------------------------------------- one GRU step (t = 1..3)
__global__ void gru_step_kernel(float* __restrict__ s_all, const float* __restrict__ Wih,
                                const float* __restrict__ bih, const float* __restrict__ gh,
                                const float* __restrict__ x, int t) {
    int idx = blockIdx.x * blockDim.x + threadIdx.x;  // b*256 + d
    int b = idx >> 8, d = idx & 255;
    const float* sp = s_all + ((size_t)(t - 1) * B_ + b) * D2_;
    const float* wr = Wih + (size_t)d * D2_;
    const float* wz = Wih + (size_t)(D2_ + d) * D2_;
    const float* wn = Wih + (size_t)(2 * D2_ + d) * D2_;
    float gir = bih[d], giz = bih[D2_ + d], gin = bih[2 * D2_ + d];
    for (int e = 0; e < D2_; ++e) {
        float sv = sp[e];
        gir += sv * wr[e];
        giz += sv * wz[e];
        gin += sv * wn[e];
    }
    const float* ghb = gh + (size_t)b * 3 * D2_;
    float r  = 1.f / (1.f + expf(-(gir + ghb[d])));
    float z  = 1.f / (1.f + expf(-(giz + ghb[D2_ + d])));
    float nn = tanhf(gin + r * ghb[2 * D2_ + d]);
    float xv = x[(size_t)b * D2_ + d];
    s_all[((size_t)t * B_ + b) * D2_ + d] = (1.f - z) * nn + z * xv;
}

// ---------------------------------------------------------------- zero the pad rows of Apad
// Apad layout: [B_][MPAD][D2_]; rows m in [T_, MPAD) must be zero.
__global__ void pad_zero_kernel(float* __restrict__ Apad) {
    int idx = blockIdx.x * blockDim.x + threadIdx.x;     // over B_*(MPAD-T_)*D2_
    int e = idx & 255;
    int r = idx >> 8;                                    // b*(MPAD-T_) + (m-T_)
    int b = r / (MPAD - T_);
    int m = T_ + (r % (MPAD - T_));
    if (b < B_) Apad[((size_t)b * MPAD + m) * D2_ + e] = 0.f;
}

// ---------------------------------------------------------------- Apad[b][t][:] = S[t*B+b][:] @ W
__global__ void proj_kernel(const float* __restrict__ S, const float* __restrict__ W,
                            float* __restrict__ Apad) {
    int idx = blockIdx.x * blockDim.x + threadIdx.x;  // r*256 + e, r = t*B_+b
    int r = idx >> 8, e = idx & 255;
    int t = r >> 5, b = r & 31;
    const float* sr = S + (size_t)r * D2_;
    float acc = 0.f;
    for (int d = 0; d < D2_; ++d) acc += sr[d] * W[(size_t)d * D2_ + e];
    Apad[((size_t)b * MPAD + t) * D2_ + e] = acc;
}

// ---------------------------------------------------------------- Apad[b][t][:] = [S,C][t*B+b] @ W_7
__global__ void proj2_kernel(const float* __restrict__ S, const float* __restrict__ C,
                             const float* __restrict__ W7, float* __restrict__ Apad) {
    int idx = blockIdx.x * blockDim.x + threadIdx.x;
    int r = idx >> 8, e = idx & 255;
    int t = r >> 5, b = r & 31;
    const float* sr = S + (size_t)r * D2_;
    const float* cr = C + (size_t)r * D2_;
    float acc = 0.f;
    for (int d = 0; d < D2_; ++d) acc += sr[d] * W7[(size_t)d * D2_ + e];
    for (int d = 0; d < D2_; ++d) acc += cr[d] * W7[(size_t)(D2_ + d) * D2_ + e];
    Apad[((size_t)b * MPAD + t) * D2_ + e] = acc;
}

// ---------------------------------------------------------------- logits[t,b,n] = Apad[b,t,:] . M[b,:,n]
// Per batch b: (16x256, rows 4..15 zero) @ (256x4096) via V_WMMA_F32_16X16X4_F32.
// A 16x4 f32 frag: lanes 0-15 carry K=0,1 ; lanes 16-31 carry K=2,3 (rows M = lane&15).
// A per-lane pair is contiguous -> single b64 load, no EXEC divergence.
// B frag mirrors the K-half split; lane's column = n0 + (lane&15).
// D: VGPR v holds row v + 8*(lane>=16), col n0 + (lane&15); only rows 0..3 stored.
__global__ void wmma_logits_kernel(const float* __restrict__ Apad,  // [B_][MPAD][D2_]
                                   const float* __restrict__ M,     // [B_, 256, 4096]
                                   float* __restrict__ out) {       // [T_*B_, 4096]
    int b    = blockIdx.x;
    int wave = threadIdx.x >> 5;
    int lane = threadIdx.x & 31;
    int n0   = (blockIdx.y * 8 + wave) * 16;
    int mr   = lane & 15;
    int kh   = (lane >> 4) << 1;                  // 0 or 2
    const float* Mb   = M + (size_t)b * D2_ * PL_;
    const float* arow = Apad + ((size_t)b * MPAD + mr) * D2_;  // rows >= T_ are zero
    int  nc = n0 + mr;                            // this lane's B column
    v8f acc = {0.f, 0.f, 0.f, 0.f, 0.f, 0.f, 0.f, 0.f};
    #pragma unroll 4
    for (int k = 0; k < D2_; k += 4) {
        v2f av = *(const v2f*)(arow + k + kh);    // unconditional b64 load
        v2f bv;
        bv.x = Mb[(size_t)(k + kh)     * PL_ + nc];
        bv.y = Mb[(size_t)(k + kh + 1) * PL_ + nc];
        acc = __builtin_amdgcn_wmma_f32_16x16x4_f32(
            /*neg_a=*/false, av, /*neg_b=*/false, bv,
            /*c_mod=*/(short)0, acc, /*reuse_a=*/false, /*reuse_b=*/false);
    }
    int col   = n0 + (lane & 15);
    int rbase = (lane >> 4) * 8;
    #pragma unroll
    for (int v = 0; v < 8; ++v) {
        int m = rbase + v;
        if (m < T_) out[((size_t)m * B_ + b) * PL_ + col] = acc[v];
    }
}

// ---------------------------------------------------------------- in-place row softmax over 4096
__global__ void softmax_rows_kernel(float* __restrict__ data) {
    int row = blockIdx.x;
    float* p = data + (size_t)row * PL_;
    __shared__ float red[256];
    int tid = threadIdx.x;
    float m = -1e30f;
    for (int n = tid; n < PL_; n += 256) m = fmaxf(m, p[n]);
    red[tid] = m; __syncthreads();
    for (int s = 128; s > 0; s >>= 1) {
        if (tid < s) red[tid] = fmaxf(red[tid], red[tid + s]);
        __syncthreads();
    }
    m = red[0]; __syncthreads();
    float acc = 0.f;
    for (int n = tid; n < PL_; n += 256) { float e = expf(p[n] - m); p[n] = e; acc += e; }
    float s = block_reduce_sum(acc, red);
    float inv = 1.f / s;
    for (int n = tid; n < PL_; n += 256) p[n] *= inv;
}

// ---------------------------------------------------------------- ctx[t,b,d] = sum_n p1[t,b,n] M[b,d,n]
__global__ void ctx_kernel(const float* __restrict__ M, const float* __restrict__ p1,
                           float* __restrict__ ctx) {
    int row = blockIdx.x;                 // b*256 + d
    int b = row >> 8, d = row & 255;
    const float* Mr = M + (size_t)row * PL_;
    const float* q0 = p1 + ((size_t)0 * B_ + b) * PL_;
    const float* q1 = p1 + ((size_t)1 * B_ + b) * PL_;
    const float* q2 = p1 + ((size_t)2 * B_ + b) * PL_;
    const float* q3 = p1 + ((size_t)3 * B_ + b) * PL_;
    float a0 = 0.f, a1 = 0.f, a2 = 0.f, a3 = 0.f;
    for (int n = threadIdx.x; n < PL_; n += 256) {
        float mvv = Mr[n];
        a0 += q0[n] * mvv; a1 += q1[n] * mvv; a2 += q2[n] * mvv; a3 += q3[n] * mvv;
    }
    __shared__ float red[256];
    float s0 = block_reduce_sum(a0, red);
    float s1 = block_reduce_sum(a1, red);
    float s2 = block_reduce_sum(a2, red);
    float s3 = block_reduce_sum(a3, red);
    if (threadIdx.x == 0) {
        ctx[((size_t)0 * B_ + b) * D2_ + d] = s0;
        ctx[((size_t)1 * B_ + b) * D2_ + d] = s1;
        ctx[((size_t)2 * B_ + b) * D2_ + d] = s2;
        ctx[((size_t)3 * B_ + b) * D2_ + d] = s3;
    }
}

// ---------------------------------------------------------------- outputs: log(sum_t p / PL)
__global__ void final_kernel(const float* __restrict__ p1, const float* __restrict__ p2,
                             float* __restrict__ out) {
    int idx = blockIdx.x * blockDim.x + threadIdx.x;  // b*4096 + n
    if (idx >= B_ * PL_) return;
    int b = idx >> 12, n = idx & 4095;
    float a = 0.f, c = 0.f;
    for (int t = 0; t < T_; ++t) {
        a += p1[((size_t)t * B_ + b) * PL_ + n];
        c += p2[((size_t)t * B_ + b) * PL_ + n];
    }
    out[idx]                    = logf(a * (1.0f / PL_));
    out[(size_t)B_ * PL_ + idx] = logf(c * (1.0f / PL_));
}

// ================================================================ launch
extern "C" void kernel_launch(void* const* d_in, const int* in_sizes, int n_in,
                              void* d_out, int out_size, void* d_ws, size_t ws_size,
                              hipStream_t stream) {
    (void)in_sizes; (void)n_in; (void)out_size; (void)ws_size;
    // inputs (H_p = d_in[0] and W_5 = d_in[4] are mathematically unused)
    const float* Hq  = (const float*)d_in[1];
    const float* M   = (const float*)d_in[2];
    const float* W4  = (const float*)d_in[3];
    const float* W6  = (const float*)d_in[5];
    const float* W7  = (const float*)d_in[6];
    const float* Wih = (const float*)d_in[7];
    const float* Whh = (const float*)d_in[8];
    const float* bih = (const float*)d_in[9];
    const float* bhh = (const float*)d_in[10];

    float* ws    = (float*)d_ws;
    float* s_all = ws;                  // T*B*D2        =  32768
    float* x     = s_all + 32768;       // B*D2          =   8192
    float* gh    = x     + 8192;        // B*3*D2        =  24576
    float* Apad  = gh    + 24576;       // B*MPAD*D2     = 131072
    float* ctx   = Apad  + 131072;      // T*B*D2        =  32768
    float* p1    = ctx   + 32768;       // T*B*PL        = 524288
    float* p2    = p1    + 524288;      // T*B*PL        = 524288
    float* out   = (float*)d_out;

    attn_s0_kernel<<<B_, 256, 0, stream>>>(Hq, W4, s_all);
    mean_x_kernel<<<B_ * D2_, 256, 0, stream>>>(M, x);
    gh_kernel<<<(B_ * 3 * D2_ + 255) / 256, 256, 0, stream>>>(x, Whh, bhh, gh);
    for (int t = 1; t < T_; ++t)
        gru_step_kernel<<<(B_ * D2_) / 256, 256, 0, stream>>>(s_all, Wih, bih, gh, x, t);

    // zero the pad rows of Apad once per launch (rows 0..3 overwritten by proj kernels)
    pad_zero_kernel<<<(B_ * (MPAD - T_) * D2_ + 255) / 256, 256, 0, stream>>>(Apad);

    // pass 1: logits1 = (s_all @ W_6) @ M  -> softmax -> p1
    proj_kernel<<<(T_ * B_ * D2_) / 256, 256, 0, stream>>>(s_all, W6, Apad);
    wmma_logits_kernel<<<dim3(B_, PL_ / 128), 256, 0, stream>>>(Apad, M, p1);
    softmax_rows_kernel<<<T_ * B_, 256, 0, stream>>>(p1);

    // pass 2: ctx = p1 @ M^T ; w = [s_all, ctx] @ W_7 ; logits2 = w @ M -> p2
    ctx_kernel<<<B_ * D2_, 256, 0, stream>>>(M, p1, ctx);
    proj2_kernel<<<(T_ * B_ * D2_) / 256, 256, 0, stream>>>(s_all, ctx, W7, Apad);
    wmma_logits_kernel<<<dim3(B_, PL_ / 128), 256, 0, stream>>>(Apad, M, p2);
    softmax_rows_kernel<<<T_ * B_, 256, 0, stream>>>(p2);

    final_kernel<<<(B_ * PL_ + 255) / 256, 256, 0, stream>>>(p1, p2, out);
}